// Loss_yolov1_28166395527878
// MI455X (gfx1250) — compile-verified
//
#include <hip/hip_runtime.h>
#include <hip/hip_bf16.h>

// ---------------------------------------------------------------------------
// YOLOv1 loss on MI455X (gfx1250, wave32).
// Memory-bound streaming reduction: ~193 MB read once -> scalar.
// - 1 thread per grid cell; channel loads are lane-contiguous (coalesced).
// - Non-temporal loads (data ~= L2 size, single touch).
// - Wave reduction via v_wmma_f32_16x16x4_f32 (B = ones => row sums),
//   then one shfl_xor(16); cross-wave via LDS. Fully deterministic.
// ---------------------------------------------------------------------------

typedef __attribute__((ext_vector_type(2))) float v2f;
typedef __attribute__((ext_vector_type(8))) float v8f;

// Full 32-lane sum, result broadcast to all lanes.
// A-matrix (16x4 f32) layout: lane L holds A[L%16][K = (L<16 ? 0 : 2)] in VGPR0
// and K+1 in VGPR1. With B = ones, D[m][n] = sum_k A[m][k] = x[m] + x[m+16].
// D VGPR j: lanes 0-15 -> row j, lanes 16-31 -> row j+8. Summing the 8 D
// components gives half-sum per lane-half; one xor-16 shuffle completes it.
__device__ __forceinline__ float wave_reduce_wmma(float x) {
    v2f a;
    a.x = x;
    a.y = 0.0f;
    v2f bones;
    bones.x = 1.0f;
    bones.y = 1.0f;
    v8f c = {};
    v8f d = __builtin_amdgcn_wmma_f32_16x16x4_f32(
        /*neg_a=*/false, a, /*neg_b=*/false, bones,
        /*c_mod=*/(short)0, c, /*reuse_a=*/false, /*reuse_b=*/false);
    float s = ((d[0] + d[1]) + (d[2] + d[3])) + ((d[4] + d[5]) + (d[6] + d[7]));
    s += __shfl_xor(s, 16, 32);
    return s;
}

__device__ __forceinline__ float sq(float v) { return v * v; }

struct Box { float x1, y1, x2, y2; };

__device__ __forceinline__ Box mkbox(float px, float py, float w, float h,
                                     float gx, float gy) {
    float cx = (px + gx) / 7.0f;
    float cy = (py + gy) / 7.0f;
    Box r;
    r.x1 = cx - w * 0.5f;
    r.y1 = cy - h * 0.5f;
    r.x2 = cx + w * 0.5f;
    r.y2 = cy + h * 0.5f;
    return r;
}

__device__ __forceinline__ float iou(const Box& a, const Box& b) {
    float x1 = fmaxf(a.x1, b.x1);
    float y1 = fmaxf(a.y1, b.y1);
    float x2 = fminf(a.x2, b.x2);
    float y2 = fminf(a.y2, b.y2);
    float inter = fmaxf(x2 - x1, 0.0f) * fmaxf(y2 - y1, 0.0f);
    float area_a = (a.x2 - a.x1) * (a.y2 - a.y1);
    float area_b = (b.x2 - b.x1) * (b.y2 - b.y1);
    return inter > 0.0f ? inter / (area_a + area_b - inter) : 0.0f;
}

__device__ __forceinline__ float ntload(const float* p) {
    return __builtin_nontemporal_load(p);
}

// ---------------------------------------------------------------------------
// Kernel 1: per-cell loss -> per-block partial sums
// ---------------------------------------------------------------------------
__global__ __launch_bounds__(256) void yolov1_loss_partial(
    const float* __restrict__ pred, const float* __restrict__ labels,
    float* __restrict__ partials, int n_cells) {
    const int cell = blockIdx.x * 256 + threadIdx.x;

    float L = 0.0f;
    if (cell < n_cells) {
        const int b = cell / 49;
        const int mn = cell - b * 49;
        const int m = mn / 7;          // dim -2 index -> gx
        const int n = mn - m * 7;      // dim -1 index -> gy
        const float gx = (float)m;
        const float gy = (float)n;

        const float* pp = pred + (long)b * (30 * 49) + mn;    // channel stride 49
        const float* lp = labels + (long)b * (30 * 49) + mn;

        // pred channels 0..9
        float p0 = ntload(pp + 0 * 49), p1 = ntload(pp + 1 * 49);
        float p2 = ntload(pp + 2 * 49), p3 = ntload(pp + 3 * 49);
        float p4 = ntload(pp + 4 * 49);
        float p5 = ntload(pp + 5 * 49), p6 = ntload(pp + 6 * 49);
        float p7 = ntload(pp + 7 * 49), p8 = ntload(pp + 8 * 49);
        float p9 = ntload(pp + 9 * 49);
        // label channels 0..8 (9 unused by the reference)
        float l0 = ntload(lp + 0 * 49), l1 = ntload(lp + 1 * 49);
        float l2 = ntload(lp + 2 * 49), l3 = ntload(lp + 3 * 49);
        float l4 = ntload(lp + 4 * 49);
        float l5 = ntload(lp + 5 * 49), l6 = ntload(lp + 6 * 49);
        float l7 = ntload(lp + 7 * 49), l8 = ntload(lp + 8 * 49);

        Box b1 = mkbox(p0, p1, p2, p3, gx, gy);
        Box b2 = mkbox(p5, p6, p7, p8, gx, gy);
        Box bg = mkbox(l0, l1, l2, l3, gx, gy);

        float iou1 = iou(b1, bg);
        float iou2 = iou(b2, bg);
        bool resp1 = iou1 > iou2;

        float obj = (l4 == 1.0f) ? 1.0f : 0.0f;

        float coor1 = sq(p0 - l0) + sq(p1 - l1) +
                      sq(sqrtf(p2) - sqrtf(l2)) + sq(sqrtf(p3) - sqrtf(l3));
        float coor2 = sq(p5 - l5) + sq(p6 - l6) +
                      sq(sqrtf(p7) - sqrtf(l7)) + sq(sqrtf(p8) - sqrtf(l8));
        float coor = resp1 ? coor1 : coor2;

        float e1 = sq(p4 - iou1);
        float e2 = sq(p9 - iou2);
        float obj_conf = resp1 ? e1 : e2;
        float noobj_resp = resp1 ? e2 : e1;

        float cls = 0.0f;
#pragma unroll
        for (int ch = 10; ch < 30; ++ch) {
            float pc = ntload(pp + ch * 49);
            float lc = ntload(lp + ch * 49);
            cls += sq(pc - lc);
        }

        L = (5.0f * coor + obj_conf + 0.5f * noobj_resp + cls) * obj +
            0.5f * (p4 * p4 + p9 * p9) * (1.0f - obj);
    }

    // Wave reduce (all 256 threads converged here -> EXEC all ones for WMMA)
    float ws = wave_reduce_wmma(L);

    __shared__ float sm[8];
    const int lane = threadIdx.x & 31;
    const int wid = threadIdx.x >> 5;
    if (lane == 0) sm[wid] = ws;
    __syncthreads();
    if (threadIdx.x == 0) {
        float t = ((sm[0] + sm[1]) + (sm[2] + sm[3])) +
                  ((sm[4] + sm[5]) + (sm[6] + sm[7]));
        partials[blockIdx.x] = t;
    }
}

// ---------------------------------------------------------------------------
// Kernel 2: sum the per-block partials (fixed order -> deterministic), /6
// ---------------------------------------------------------------------------
__global__ __launch_bounds__(256) void yolov1_loss_finish(
    const float* __restrict__ partials, int n, float* __restrict__ out) {
    float s = 0.0f;
    for (int i = threadIdx.x; i < n; i += 256) s += partials[i];

    float ws = wave_reduce_wmma(s);  // all threads reconverged before this

    __shared__ float sm[8];
    const int lane = threadIdx.x & 31;
    const int wid = threadIdx.x >> 5;
    if (lane == 0) sm[wid] = ws;
    __syncthreads();
    if (threadIdx.x == 0) {
        float t = ((sm[0] + sm[1]) + (sm[2] + sm[3])) +
                  ((sm[4] + sm[5]) + (sm[6] + sm[7]));
        // Reference faithfully divides by range(7)[-1] == 6 (shadowed n bug).
        out[0] = t / 6.0f;
    }
}

extern "C" void kernel_launch(void* const* d_in, const int* in_sizes, int n_in,
                              void* d_out, int out_size, void* d_ws, size_t ws_size,
                              hipStream_t stream) {
    const float* pred = (const float*)d_in[0];
    const float* labels = (const float*)d_in[1];

    const int C = 30, GG = 49;
    const int B = in_sizes[0] / (C * GG);   // 16384
    const int n_cells = B * GG;             // 802816
    const int blocks = (n_cells + 255) / 256;  // 3136

    float* partials = (float*)d_ws;  // blocks * 4 bytes (~12.5 KB) << ws_size

    yolov1_loss_partial<<<blocks, 256, 0, stream>>>(pred, labels, partials, n_cells);
    yolov1_loss_finish<<<1, 256, 0, stream>>>(partials, blocks, (float*)d_out);
}